// SelfIntersectionsCPU_81750407512649
// MI455X (gfx1250) — compile-verified
//
#include <hip/hip_runtime.h>

typedef float v2f __attribute__((ext_vector_type(2)));
typedef float v8f __attribute__((ext_vector_type(8)));

#define TTI_EPS 1e-6f

__device__ __forceinline__ void interval3(const float p[3], const float d[3],
                                          float& tmin, float& tmax) {
  tmin = INFINITY;
  tmax = -INFINITY;
#pragma unroll
  for (int e = 0; e < 3; ++e) {
    const int i = e;
    const int j = (e + 1) % 3;
    const float di = d[i], dj = d[j];
    const bool valid = (di * dj <= 0.0f) && !((di == 0.0f) && (dj == 0.0f));
    const float denom = di - dj;
    const float safe = (denom == 0.0f) ? 1.0f : denom;
    const float t = p[i] + (p[j] - p[i]) * di / safe;
    if (valid) {
      tmin = fminf(tmin, t);
      tmax = fmaxf(tmax, t);
    }
  }
}

__global__ __launch_bounds__(256) void tritri_pairs_kernel(
    const float* __restrict__ T, float* __restrict__ out, int N) {
  // LDS staging: vertex quads (x,y,z,1), plane quads (nx,ny,nz,d),
  // and the two 48x16 signed-distance blocks produced by WMMA.
  __shared__ float Vb4[48 * 4];   // row-tile vertices
  __shared__ float Vu4[48 * 4];   // col-tile vertices
  __shared__ float Pb4[16 * 4];   // row-tile planes
  __shared__ float Pu4[16 * 4];   // col-tile planes
  __shared__ float dvs[48 * 16];  // dist(b-verts, u-planes)
  __shared__ float dus[48 * 16];  // dist(u-verts, b-planes)

  const int tid = threadIdx.x;
  const int b0 = blockIdx.y * 16;  // row tile (first index of pair matrix)
  const int u0 = blockIdx.x * 16;  // col tile

  // ---- stage vertices + planes into LDS ----
  if (tid < 48) {
    const int tri = b0 + tid / 3, v = tid % 3;
    const float* p = T + ((size_t)tri * 3 + v) * 3;
    Vb4[tid * 4 + 0] = p[0];
    Vb4[tid * 4 + 1] = p[1];
    Vb4[tid * 4 + 2] = p[2];
    Vb4[tid * 4 + 3] = 1.0f;
  } else if (tid < 96) {
    const int k = tid - 48;
    const int tri = u0 + k / 3, v = k % 3;
    const float* p = T + ((size_t)tri * 3 + v) * 3;
    Vu4[k * 4 + 0] = p[0];
    Vu4[k * 4 + 1] = p[1];
    Vu4[k * 4 + 2] = p[2];
    Vu4[k * 4 + 3] = 1.0f;
  } else if (tid < 128) {
    const int i = tid - 96;
    const int tri = (i < 16) ? (b0 + i) : (u0 + (i - 16));
    const float* p = T + (size_t)tri * 9;
    const float e1x = p[3] - p[0], e1y = p[4] - p[1], e1z = p[5] - p[2];
    const float e2x = p[6] - p[0], e2y = p[7] - p[1], e2z = p[8] - p[2];
    const float nx = e1y * e2z - e1z * e2y;
    const float ny = e1z * e2x - e1x * e2z;
    const float nz = e1x * e2y - e1y * e2x;
    const float d = -(nx * p[0] + ny * p[1] + nz * p[2]);
    float* dst = (i < 16) ? (Pb4 + i * 4) : (Pu4 + (i - 16) * 4);
    dst[0] = nx;
    dst[1] = ny;
    dst[2] = nz;
    dst[3] = d;
  }
  __syncthreads();

  // ---- WMMA distance blocks: D = Verts(16x4) x Planes^T(4x16) ----
  // waves 0..2 -> dvs vertex groups 0..2 ; waves 3..5 -> dus groups 0..2
  const int wave = tid >> 5;
  const int lane = tid & 31;
  if (wave < 6) {
    const int grp = (wave % 3) * 16;       // vertex-row group base (of 48)
    const int m = lane & 15;               // M (A) / N (B) index
    const int kh = (lane >> 4) << 1;       // K half: 0 or 2
    const float* V = (wave < 3) ? Vb4 : Vu4;
    const float* P = (wave < 3) ? Pu4 : Pb4;
    float* Dst = (wave < 3) ? dvs : dus;

    v2f a, b;
    a.x = V[(grp + m) * 4 + kh + 0];  // A: lane half holds K={0,1} or {2,3}
    a.y = V[(grp + m) * 4 + kh + 1];
    b.x = P[m * 4 + kh + 0];          // B: column m, K rows split by lane half
    b.y = P[m * 4 + kh + 1];

    v8f c = {0.f, 0.f, 0.f, 0.f, 0.f, 0.f, 0.f, 0.f};
    // 8 args: (neg_a, A, neg_b, B, c_mod, C, reuse_a, reuse_b)
    c = __builtin_amdgcn_wmma_f32_16x16x4_f32(false, a, false, b, (short)0, c,
                                              false, false);

    // C/D layout: VGPR r, lanes 0-15 -> row r ; lanes 16-31 -> row r+8
    const int rb = (lane >> 4) * 8;
#pragma unroll
    for (int r = 0; r < 8; ++r) {
      Dst[(grp + rb + r) * 16 + m] = c[r];
    }
  }
  __syncthreads();

  // ---- per-pair Moller interval test (one thread per pair) ----
  const int bi = tid >> 4;   // row-local triangle
  const int ui = tid & 15;   // col-local triangle

  float dvv[3], duu[3];
#pragma unroll
  for (int v = 0; v < 3; ++v) {
    float x = dvs[(3 * bi + v) * 16 + ui];
    dvv[v] = (fabsf(x) < TTI_EPS) ? 0.0f : x;
    float y = dus[(3 * ui + v) * 16 + bi];
    duu[v] = (fabsf(y) < TTI_EPS) ? 0.0f : y;
  }

  const bool ssv = ((dvv[0] > 0.f) && (dvv[1] > 0.f) && (dvv[2] > 0.f)) ||
                   ((dvv[0] < 0.f) && (dvv[1] < 0.f) && (dvv[2] < 0.f));
  const bool ssu = ((duu[0] > 0.f) && (duu[1] > 0.f) && (duu[2] > 0.f)) ||
                   ((duu[0] < 0.f) && (duu[1] < 0.f) && (duu[2] < 0.f));

  // direction of plane-plane intersection line = cross(n_b, n_u)
  const float nbx = Pb4[bi * 4 + 0], nby = Pb4[bi * 4 + 1], nbz = Pb4[bi * 4 + 2];
  const float nux = Pu4[ui * 4 + 0], nuy = Pu4[ui * 4 + 1], nuz = Pu4[ui * 4 + 2];
  const float Dx = nby * nuz - nbz * nuy;
  const float Dy = nbz * nux - nbx * nuz;
  const float Dz = nbx * nuy - nby * nux;

  int ax = 0;
  float best = fabsf(Dx);
  const float aDy = fabsf(Dy);
  if (aDy > best) { best = aDy; ax = 1; }
  const float aDz = fabsf(Dz);
  if (aDz > best) { ax = 2; }

  float pv[3], pu[3];
#pragma unroll
  for (int v = 0; v < 3; ++v) {
    pv[v] = Vb4[(3 * bi + v) * 4 + ax];
    pu[v] = Vu4[(3 * ui + v) * 4 + ax];
  }

  float t0v, t1v, t0u, t1u;
  interval3(pv, dvv, t0v, t1v);
  interval3(pu, duu, t0u, t1u);

  const bool overlap = fmaxf(t0v, t0u) <= fminf(t1v, t1u);
  const bool hit = (!ssv) && (!ssu) && overlap;

  const int gb = b0 + bi;
  const int gu = u0 + ui;
  if (gb < N && gu < N) {
    out[(size_t)gb * N + gu] = (gb == gu) ? 1.0f : (hit ? 1.0f : 0.0f);
  }
}

extern "C" void kernel_launch(void* const* d_in, const int* in_sizes, int n_in,
                              void* d_out, int out_size, void* d_ws, size_t ws_size,
                              hipStream_t stream) {
  (void)n_in;
  (void)out_size;
  (void)d_ws;
  (void)ws_size;
  const float* T = (const float*)d_in[0];
  float* out = (float*)d_out;
  const int N = in_sizes[0] / 9;           // [N,3,3] float32 triangle soup
  const int tiles = (N + 15) / 16;
  dim3 grid(tiles, tiles);
  dim3 block(256);
  tritri_pairs_kernel<<<grid, block, 0, stream>>>(T, out, N);
}